// Embedding_27986006901450
// MI455X (gfx1250) — compile-verified
//
#include <hip/hip_runtime.h>
#include <hip/hip_bf16.h>

typedef float v2f __attribute__((ext_vector_type(2)));
typedef float v8f __attribute__((ext_vector_type(8)));

#define B_DIM 128
#define N_DIM 1024
#define D_DIM 512

// out[b,i,d] = x[b,i] * W[i,d] + bias[i,d]
//
// One wave owns a (neuron i, 16-wide d tile) pair and loops over all 8 batch
// tiles. Per batch tile it computes the 16(batch) x 16(d) output tile with
// V_WMMA_F32_16X16X4_F32:  D = A(16x4) * B(4x16) + 0
//   A: K=0 column = x[b0..b0+15, i], K=1 column = 1.0, K=2,3 = 0
//   B: K=0 row    = W[i, d0..d0+15], K=1 row   = bias[i, d0..d0+15], K=2,3 = 0
// => D[m,n] = x[b0+m]*W[d0+n] + bias[d0+n]
//
// W/bias are loaded once per wave (L2-resident 4MB reuse set); the 256MB
// output stream is written with non-temporal stores so it does not evict them.
__global__ __launch_bounds__(256) void neuron_linear_wmma(
    const float* __restrict__ X,   // [B, N]   (x[b,i,0])
    const float* __restrict__ Wt,  // [N, D]
    const float* __restrict__ Bs,  // [N, D]
    float* __restrict__ Out)       // [B, N, D]
{
    const int wavesPerBlock = blockDim.x >> 5;          // 8 (wave32)
    const int wave = blockIdx.x * wavesPerBlock + (threadIdx.x >> 5);
    const int lane = threadIdx.x & 31;

    const int nDTiles = D_DIM / 16;                     // 32
    const int dTile = wave & (nDTiles - 1);             // adjacent waves -> adjacent d tiles
    const int i     = wave / nDTiles;                   // neuron index 0..1023

    const int d0 = dTile * 16;
    const int l  = lane & 15;
    const bool lo = lane < 16;          // lanes 0-15 carry K=0/1; lanes 16-31 are K=2/3 (zeros)

    // B operand: built once, reused for all 8 batch tiles.
    const float wv = Wt[(size_t)i * D_DIM + d0 + l];
    const float bv = Bs[(size_t)i * D_DIM + d0 + l];
    v2f Bm;
    Bm[0] = lo ? wv   : 0.0f;           // K=0 row = W      (K=2 row = 0)
    Bm[1] = lo ? bv   : 0.0f;           // K=1 row = bias   (K=3 row = 0)

    v2f A;
    A[1] = lo ? 1.0f : 0.0f;            // K=1 column = 1.0 (K=3 = 0)

    const size_t rowStride = (size_t)N_DIM * D_DIM;     // +1 batch row (in floats)
    // Column base for this wave's stores: row (lo?0:8), neuron i, col d0+l.
    float* outCol = Out + ((size_t)(lo ? 0 : 8) * N_DIM + i) * D_DIM + d0 + l;
    const float* xCol = X + i;                          // x[:, i], stride N_DIM

#pragma unroll
    for (int bT = 0; bT < B_DIM / 16; ++bT) {           // 8 batch tiles
        const int b0 = bT * 16;

        // 16-lane gather of x[b0..b0+15, i]; lanes 16-31 read a valid address
        // (same l range) then get zeroed via cndmask — no EXEC divergence.
        const float xv = xCol[(size_t)(b0 + l) * N_DIM];
        A[0] = lo ? xv : 0.0f;          // K=0 column = x   (K=2 = 0)

        v8f C = {};
        // (neg_a, A, neg_b, B, c_mod, C, reuse_a, reuse_b)
        v8f Dm = __builtin_amdgcn_wmma_f32_16x16x4_f32(
            false, A, false, Bm, (short)0, C, false, false);

        // C/D layout: VGPR v, lanes 0-15 -> (M=v, N=lane), lanes 16-31 -> (M=v+8, N=lane-16)
        float* p = outCol + (size_t)b0 * rowStride;
#pragma unroll
        for (int v = 0; v < 8; ++v) {
            __builtin_nontemporal_store(Dm[v], p + (size_t)v * rowStride);
        }
    }
}

extern "C" void kernel_launch(void* const* d_in, const int* in_sizes, int n_in,
                              void* d_out, int out_size, void* d_ws, size_t ws_size,
                              hipStream_t stream) {
    const float* X  = (const float*)d_in[0];  // [B, N, 1]
    const float* Wt = (const float*)d_in[1];  // [N, D]
    const float* Bs = (const float*)d_in[2];  // [N, D]
    float* Out = (float*)d_out;               // [B, N, D]

    const int totalWaves = N_DIM * (D_DIM / 16);        // 32768
    const int blockSize  = 256;                         // 8 waves (wave32)
    const int numBlocks  = totalWaves / (blockSize / 32); // 4096

    neuron_linear_wmma<<<numBlocks, blockSize, 0, stream>>>(X, Wt, Bs, Out);
}